// TiedStateSpaceModel_27333171872510
// MI455X (gfx1250) — compile-verified
//
#include <hip/hip_runtime.h>
#include <hip/hip_bf16.h>
#include <math.h>

// Problem constants (from reference): Q=4, N=64, H=256, C=1, L=1024
#define QD   4
#define ND   64
#define HD   256
#define LD   1024
#define QH   (QD * HD)          // 1024 columns
#define KF   513                // frequency bins used by irfft (L/2+1)
#define KDIM 1028               // 2*514 K rows (f=0..513, f=513 is zero pad)

typedef float v2f __attribute__((ext_vector_type(2)));
typedef float v8f __attribute__((ext_vector_type(8)));

// ---------------------------------------------------------------------------
// Pass 1: frequency-domain SSM kernel.
// One thread per (l, qh), l in 0..513 (513 = zero-pad pair for K padding).
// Gp layout (pair-interleaved for b64 B-fragment loads):
//   Gp[(f*1024 + qh)*2 + 0] =  alpha_f * Re(X_f)   (WMMA K row 2f)
//   Gp[(f*1024 + qh)*2 + 1] = -alpha_f * Im(X_f)   (WMMA K row 2f+1)
// ---------------------------------------------------------------------------
__global__ __launch_bounds__(256)
void ssm_freq_kernel(const float* __restrict__ diagonal,      // (Q,N)
                     const float* __restrict__ lowrank,       // (Q,N)
                     const float* __restrict__ timestep,      // (Q,H)
                     const float* __restrict__ input_matrix,  // (Q,N)
                     const float* __restrict__ output_matrix, // (Q,1,H,N)
                     float* __restrict__ Gp)                  // (514,1024,2)
{
    const int idx = blockIdx.x * 256 + threadIdx.x;
    const int l   = idx >> 10;        // block-uniform (256 | 1024)
    const int qh  = idx & 1023;

    if (l == KF) {                    // zero pad pair (K rows 1026/1027)
        Gp[(KF * QH + qh) * 2 + 0] = 0.0f;
        Gp[(KF * QH + qh) * 2 + 1] = 0.0f;
        return;
    }

    const int q = qh >> 8;            // block-uniform
    const int h = qh & 255;

    // Stage per-q vectors in LDS once per block.
    __shared__ float s_d[ND];         // d_n = -exp(diagonal)
    __shared__ float s_B[ND];
    __shared__ float s_P[ND];
    __shared__ float s_w01[ND];       // B*P
    __shared__ float s_w11[ND];       // P*P
    if (threadIdx.x < ND) {
        const int n = threadIdx.x;
        float dg = diagonal[q * ND + n];
        float Bn = input_matrix[q * ND + n];
        float Pn = lowrank[q * ND + n];
        s_d[n]   = -expf(dg);
        s_B[n]   = Bn;
        s_P[n]   = Pn;
        s_w01[n] = Bn * Pn;
        s_w11[n] = Pn * Pn;
    }
    __syncthreads();

    const float t  = expf(timestep[q * HD + h]);
    const float it = 1.0f / t;

    // omega = exp(-i*theta) = c - i*s, theta = 2*pi*l/1024
    const float theta = (float)l * (6.28318530717958647692f / 1024.0f);
    float s, c;
    __sincosf(theta, &s, &c);

    // z = 2*(1-omega)/(1+omega)
    const float nr = 1.0f - c, ni = s;       // 1 - omega
    const float dr = 1.0f + c, di = -s;      // 1 + omega
    const float idn = 1.0f / (dr * dr + di * di);
    const float zr = 2.0f * (nr * dr + ni * di) * idn;
    const float zi = 2.0f * (ni * dr - nr * di) * idn;
    const float ztr = zr * it;
    const float zti = zi * it;

    // unit = 2/(1+omega) = 2*((1+c) + i*s)/den
    const float ur = 2.0f * dr * idn;
    const float ui = -2.0f * di * idn;

    const float* __restrict__ Crow = output_matrix + (q * HD + h) * ND;

    float k00r = 0.f, k00i = 0.f, k10r = 0.f, k10i = 0.f;
    float k01r = 0.f, k01i = 0.f, k11r = 0.f, k11i = 0.f;

#pragma unroll 8
    for (int n = 0; n < ND; ++n) {
        // r_n = 1/(zt - d_n), d_n real => conj over |.|^2
        const float er  = ztr - s_d[n];
        const float inv = 1.0f / (er * er + zti * zti);
        const float rr  = er * inv;
        const float ri  = -zti * inv;

        const float Cn  = Crow[n];
        const float w00 = s_B[n] * Cn;
        const float w10 = s_P[n] * Cn;
        const float w01 = s_w01[n];
        const float w11 = s_w11[n];

        k00r = fmaf(w00, rr, k00r);  k00i = fmaf(w00, ri, k00i);
        k10r = fmaf(w10, rr, k10r);  k10i = fmaf(w10, ri, k10i);
        k01r = fmaf(w01, rr, k01r);  k01i = fmaf(w01, ri, k01i);
        k11r = fmaf(w11, rr, k11r);  k11i = fmaf(w11, ri, k11i);
    }

    // kernel = k00 - k10*k01/(1 + k11)   (complex)
    const float pr   = k10r * k01r - k10i * k01i;
    const float pi   = k10r * k01i + k10i * k01r;
    const float er2  = 1.0f + k11r;
    const float ei2  = k11i;
    const float inv2 = 1.0f / (er2 * er2 + ei2 * ei2);
    const float qr   = (pr * er2 + pi * ei2) * inv2;
    const float qi   = (pi * er2 - pr * ei2) * inv2;
    const float kr   = k00r - qr;
    const float ki   = k00i - qi;

    // X = unit * kernel
    const float Xr = kr * ur - ki * ui;
    const float Xi = kr * ui + ki * ur;

    // irfft folding: alpha = 1/L for DC & Nyquist, 2/L otherwise; -Im sign
    const float alpha = (l == 0 || l == (LD / 2)) ? (1.0f / 1024.0f)
                                                  : (2.0f / 1024.0f);
    Gp[(l * QH + qh) * 2 + 0] =  alpha * Xr;
    Gp[(l * QH + qh) * 2 + 1] = -alpha * Xi;
}

// ---------------------------------------------------------------------------
// Pass 2: irfft as an f32 WMMA GEMM.
//   out[t, qh] = sum_{f=0}^{512} cos(2pi*f*t/1024)*Gre[f,qh]
//                               + sin(2pi*f*t/1024)*Gim[f,qh]
// M = t (1024), N = qh (1024), K = 1028.  Each wave computes a 16x64 strip
// (1 M-tile x 4 N-tiles): one sincos A-fragment per K-step feeds 4
// back-to-back V_WMMA_F32_16X16X4_F32 with independent accumulators, and
// each B fragment is a single b64 load from the pair-interleaved Gp.
// ---------------------------------------------------------------------------
__global__ __launch_bounds__(256)
void ssm_irfft_wmma_kernel(const float* __restrict__ Gp,  // (514,1024,2)
                           float* __restrict__ out)       // (1024,1024)
{
    const int lane = threadIdx.x & 31;
    const int wave = threadIdx.x >> 5;
    const int wid  = blockIdx.x * 8 + wave;   // 0..1023
    const int tm   = wid >> 4;                // M tile (0..63)
    const int tg   = wid & 15;                // N group of 4 tiles (0..15)
    const int m0   = tm * 16;
    const int n0   = tg * 64;

    const int hi  = lane >> 4;                // 0: lanes 0-15, 1: lanes 16-31
    const int l15 = lane & 15;
    const int t   = m0 + l15;                 // A-matrix row (time index)
    const int c0  = n0 + l15;                 // first B column for this lane

    v8f acc0 = {}, acc1 = {}, acc2 = {}, acc3 = {};
    const float ascale = 6.28318530717958647692f / 1024.0f;

#pragma unroll 4
    for (int k0 = 0; k0 < KDIM; k0 += 4) {
        // A fragment (16x4 f32): lane half selects K pair {k0,k0+1}/{k0+2,k0+3}
        const int f = (k0 >> 1) + hi;         // frequency index (even K /2)
        v2f a;
        if (f < KF) {
            const int   ph  = (f * t) & 1023; // exact phase reduction
            const float ang = (float)ph * ascale;
            float sv, cv;
            __sincosf(ang, &sv, &cv);
            a.x = cv;                         // K row 2f   (cos)
            a.y = sv;                         // K row 2f+1 (sin)
        } else {
            a.x = 0.0f;
            a.y = 0.0f;
        }
        // B fragments: one b64 load per N tile (re,im pair for frequency f)
        const float* __restrict__ base = Gp + (size_t)f * (QH * 2);
        v2f b0 = *(const v2f*)(base + (c0 +  0) * 2);
        v2f b1 = *(const v2f*)(base + (c0 + 16) * 2);
        v2f b2 = *(const v2f*)(base + (c0 + 32) * 2);
        v2f b3 = *(const v2f*)(base + (c0 + 48) * 2);

        acc0 = __builtin_amdgcn_wmma_f32_16x16x4_f32(false, a, false, b0,
                                                     (short)0, acc0, false, false);
        acc1 = __builtin_amdgcn_wmma_f32_16x16x4_f32(false, a, false, b1,
                                                     (short)0, acc1, false, false);
        acc2 = __builtin_amdgcn_wmma_f32_16x16x4_f32(false, a, false, b2,
                                                     (short)0, acc2, false, false);
        acc3 = __builtin_amdgcn_wmma_f32_16x16x4_f32(false, a, false, b3,
                                                     (short)0, acc3, false, false);
    }

    // D layout: VGPR v -> row m0+v (lanes 0-15) / m0+8+v (lanes 16-31)
    const int rbase = m0 + 8 * hi;
#pragma unroll
    for (int v = 0; v < 8; ++v) {
        float* __restrict__ row = out + (size_t)(rbase + v) * QH;
        row[c0 +  0] = acc0[v];
        row[c0 + 16] = acc1[v];
        row[c0 + 32] = acc2[v];
        row[c0 + 48] = acc3[v];
    }
}

extern "C" void kernel_launch(void* const* d_in, const int* in_sizes, int n_in,
                              void* d_out, int out_size, void* d_ws, size_t ws_size,
                              hipStream_t stream) {
    const float* diagonal      = (const float*)d_in[0];
    const float* lowrank       = (const float*)d_in[1];
    const float* timestep      = (const float*)d_in[2];
    const float* input_matrix  = (const float*)d_in[3];
    const float* output_matrix = (const float*)d_in[4];
    // d_in[5] = maxlen (scalar int, always 1024) — baked into the kernels.

    float* Gp  = (float*)d_ws;   // 514*1024*2*4 = 4.2 MB of scratch
    float* out = (float*)d_out;  // (1024, 4, 1, 256) f32

    // Pass 1: 514 l-slices x 1024 qh, 256 threads/block (8 waves)
    ssm_freq_kernel<<<(514 * 1024) / 256, 256, 0, stream>>>(
        diagonal, lowrank, timestep, input_matrix, output_matrix, Gp);

    // Pass 2: 1024 waves (64 M-tiles x 16 N-groups), 8 waves per block
    ssm_irfft_wmma_kernel<<<1024 / 8, 256, 0, stream>>>(Gp, out);
}